// FCSVec_4810363372657
// MI455X (gfx1250) — compile-verified
//
#include <hip/hip_runtime.h>

typedef float v2f __attribute__((ext_vector_type(2)));
typedef float v8f __attribute__((ext_vector_type(8)));

#define Bn 4
#define Qn 128
#define Kn 128
#define Dn 3
#define Cn 256

// ---------------------------------------------------------------------------
// Stage 1: h[b,k,:] = silu(x[b,k,:] @ W1 + b1) @ W2 + b2
// One 256-thread block per (b,k) row; weights stay hot in the 192MB L2.
// ---------------------------------------------------------------------------
__global__ __launch_bounds__(256) void mlp_kernel(
    const float* __restrict__ x,
    const float* __restrict__ W1, const float* __restrict__ b1,
    const float* __restrict__ W2, const float* __restrict__ b2,
    float* __restrict__ h)
{
    __shared__ float xs[Cn];
    __shared__ float hs[Cn];
    const int row = blockIdx.x;          // b*K + k
    const int t   = threadIdx.x;         // output channel

    xs[t] = x[(size_t)row * Cn + t];
    __syncthreads();

    float acc = b1[t];
    #pragma unroll 8
    for (int c = 0; c < Cn; ++c)
        acc = fmaf(xs[c], W1[(size_t)c * Cn + t], acc);

    // SiLU
    acc = acc / (1.0f + __expf(-acc));
    hs[t] = acc;
    __syncthreads();

    float acc2 = b2[t];
    #pragma unroll 8
    for (int c = 0; c < Cn; ++c)
        acc2 = fmaf(hs[c], W2[(size_t)c * Cn + t], acc2);

    h[(size_t)row * Cn + t] = acc2;
}

// ---------------------------------------------------------------------------
// Stage 2: out[b,q,d,c] = sum_k (mask*ev)[d,k] * (ef*h)[k,c]
// Per (b,q): (3x128)x(128x256) GEMM, D padded to M=16 for WMMA f32 16x16x4.
// One 512-thread block (16 waves) per (b,q); wave w owns C columns [16w,16w+16).
// ---------------------------------------------------------------------------
__global__ __launch_bounds__(512) void contract_kernel(
    const float* __restrict__ ev, const float* __restrict__ ef,
    const float* __restrict__ mask, const float* __restrict__ h,
    float* __restrict__ out)
{
    __shared__ float a_s[16 * Kn];       // 8KB: A matrix, rows d=3..15 zero

    const int bq  = blockIdx.x;          // b*Q + q
    const int b   = bq / Qn;
    const int tid = threadIdx.x;

    // Build A = mask * ev once per block (shared by all 16 waves).
    for (int i = tid; i < 16 * Kn; i += 512) {
        float v = 0.0f;
        if (i < Dn * Kn) {
            const int d = i >> 7;        // row (0..2)
            const int k = i & (Kn - 1);
            v = mask[(size_t)bq * Kn + k] * ev[((size_t)bq * Kn + k) * Dn + d];
        }
        a_s[i] = v;
    }
    __syncthreads();

    const int wave = tid >> 5;           // 0..15 -> C tile
    const int lane = tid & 31;
    const int m    = lane & 15;
    const int hi   = lane >> 4;
    const int c0   = wave * 16;

    const float* efp = ef  + (size_t)bq * Kn * Cn + c0 + m;  // stride Cn per k
    const float* hp  = h   + (size_t)b  * Kn * Cn + c0 + m;  // stride Cn per k
    const float* ap  = a_s + m * Kn;

    v8f acc = {0.f, 0.f, 0.f, 0.f, 0.f, 0.f, 0.f, 0.f};

    #pragma unroll 4
    for (int k4 = 0; k4 < Kn; k4 += 4) {
        const int k = k4 + 2 * hi;

        // A fragment (16x4 f32): lane m holds row m; K lo pair in lanes 0-15,
        // K hi pair in lanes 16-31; VGPR0=K even, VGPR1=K odd.
        v2f afrag;
        afrag.x = ap[k];
        afrag.y = ap[k + 1];

        // B fragment (4x16 f32): row k striped across lanes 0-15 (N=c0+m),
        // rows k4+2/k4+3 in lanes 16-31. ef streamed non-temporally (no reuse);
        // h is L2-resident.
        const float e0 = __builtin_nontemporal_load(&efp[(size_t)k * Cn]);
        const float e1 = __builtin_nontemporal_load(&efp[(size_t)(k + 1) * Cn]);
        v2f bfrag;
        bfrag.x = e0 * hp[(size_t)k * Cn];
        bfrag.y = e1 * hp[(size_t)(k + 1) * Cn];

        acc = __builtin_amdgcn_wmma_f32_16x16x4_f32(
            false, afrag, false, bfrag, (short)0, acc, false, false);
    }

    // D layout: VGPR v, lanes 0-15 -> M=v. Only M=0..2 are real output rows.
    if (lane < 16) {
        float* op = out + (size_t)bq * Dn * Cn + c0 + lane;
        op[0]        = acc[0];
        op[Cn]       = acc[1];
        op[2 * Cn]   = acc[2];
    }
}

// ---------------------------------------------------------------------------
extern "C" void kernel_launch(void* const* d_in, const int* in_sizes, int n_in,
                              void* d_out, int out_size, void* d_ws, size_t ws_size,
                              hipStream_t stream)
{
    const float* x    = (const float*)d_in[0];
    const float* ev   = (const float*)d_in[1];
    const float* ef   = (const float*)d_in[2];
    const float* mask = (const float*)d_in[3];
    const float* W1   = (const float*)d_in[4];
    const float* b1   = (const float*)d_in[5];
    const float* W2   = (const float*)d_in[6];
    const float* b2   = (const float*)d_in[7];
    float* out = (float*)d_out;
    float* h   = (float*)d_ws;                 // B*K*C floats = 512KB scratch

    mlp_kernel<<<Bn * Kn, 256, 0, stream>>>(x, W1, b1, W2, b2, h);
    contract_kernel<<<Bn * Qn, 512, 0, stream>>>(ev, ef, mask, h, out);
}